// AttentionBlock_37228776521813
// MI455X (gfx1250) — compile-verified
//
#include <hip/hip_runtime.h>
#include <hip/hip_bf16.h>
#include <math.h>

typedef __attribute__((ext_vector_type(2))) float v2f;
typedef __attribute__((ext_vector_type(8))) float v8f;

#define C_DIM 512
#define HW_DIM 1024
#define THREADS 256

// One block per batch image b.
// Pass 1: c[b,hw] = dot_C(l[b,:,hw], w) + dot(g[b,:], w)   via V_WMMA_F32_16X16X4_F32
//         (A = w[k]*l tile with documented 16x4 f32 A-layout, B = ones so
//          D[m][n] = sum_k A[m][k] independent of B's VGPR layout; accumulate over C in C/D)
// Softmax over the 1024 spatial positions in LDS.
// Pass 2: g_out[b,c] = dot_HW(l[b,c,:], a[b,:])  -- l slice is L2-resident from pass 1.
__global__ __launch_bounds__(THREADS) void attn_fused_kernel(
    const float* __restrict__ l,   // [B, C, H, W]
    const float* __restrict__ g,   // [B, C]
    const float* __restrict__ w,   // [C]
    float* __restrict__ out_c,     // [B, 1, H, W]
    float* __restrict__ out_g)     // [B, C]
{
    __shared__ float w_s[C_DIM];
    __shared__ __align__(16) float cv[HW_DIM];
    __shared__ float red[THREADS];

    const int b      = blockIdx.x;
    const int tid    = threadIdx.x;
    const int lane   = tid & 31;
    const int wv     = tid >> 5;        // wave id 0..7
    const int lane15 = lane & 15;
    const int half   = lane >> 4;       // 0 for lanes 0-15, 1 for lanes 16-31

    const float* lb = l + (size_t)b * C_DIM * HW_DIM;

    // ---- load w into LDS; compute gb = dot(g[b,:], w) ----
    float p = 0.f;
    for (int i = tid; i < C_DIM; i += THREADS) {
        float wi = w[i];
        w_s[i] = wi;
        p += wi * g[b * C_DIM + i];
    }
    red[tid] = p;
    __syncthreads();
    for (int s = 128; s > 0; s >>= 1) {
        if (tid < s) red[tid] += red[tid + s];
        __syncthreads();
    }
    const float gb = red[0];
    __syncthreads();

    // ---- pass 1: WMMA channel reduction ----
    // Wave wv owns hw range [wv*128, wv*128+128): 8 tiles of 16 positions.
    v8f acc[8];
    #pragma unroll
    for (int j = 0; j < 8; ++j)
        acc[j] = (v8f){0.f, 0.f, 0.f, 0.f, 0.f, 0.f, 0.f, 0.f};
    const v2f onesv = {1.0f, 1.0f};

    const int hwbase = wv * 128 + lane15;   // + j*16 per tile
    const int krow   = half * 2;            // A layout: VGPR0 -> K = 2*half, VGPR1 -> K = 2*half+1

    for (int c0 = 0; c0 < C_DIM; c0 += 4) {
        const float wa = w_s[c0 + krow];
        const float wb = w_s[c0 + krow + 1];
        const float* base0 = lb + (size_t)(c0 + krow) * HW_DIM + hwbase;
        const float* base1 = base0 + HW_DIM;
        #pragma unroll
        for (int j = 0; j < 8; ++j) {
            v2f A;
            A.x = wa * base0[j * 16];   // A[m][2*half]   pre-scaled by w
            A.y = wb * base1[j * 16];   // A[m][2*half+1] pre-scaled by w
            acc[j] = __builtin_amdgcn_wmma_f32_16x16x4_f32(
                false, A, false, onesv, (short)0, acc[j], false, false);
        }
    }

    // D[m][n] identical across n; VGPR r holds M=r (lanes 0-15) / M=r+8 (lanes 16-31).
    if (lane15 == 0) {
        #pragma unroll
        for (int j = 0; j < 8; ++j) {
            const int t16 = wv * 128 + j * 16 + half * 8;
            #pragma unroll
            for (int r = 0; r < 8; ++r)
                cv[t16 + r] = acc[j][r] + gb;
        }
    }
    __syncthreads();

    // ---- emit c, softmax over HW in LDS ----
    for (int i = tid; i < HW_DIM; i += THREADS)
        out_c[b * HW_DIM + i] = cv[i];

    float vloc[4];
    float m = -3.402823466e38f;
    #pragma unroll
    for (int k = 0; k < 4; ++k) {
        vloc[k] = cv[tid + k * THREADS];
        m = fmaxf(m, vloc[k]);
    }
    red[tid] = m;
    __syncthreads();
    for (int s = 128; s > 0; s >>= 1) {
        if (tid < s) red[tid] = fmaxf(red[tid], red[tid + s]);
        __syncthreads();
    }
    m = red[0];
    __syncthreads();

    float e[4];
    float ssum = 0.f;
    #pragma unroll
    for (int k = 0; k < 4; ++k) {
        e[k] = expf(vloc[k] - m);
        ssum += e[k];
    }
    red[tid] = ssum;
    __syncthreads();
    for (int s = 128; s > 0; s >>= 1) {
        if (tid < s) red[tid] += red[tid + s];
        __syncthreads();
    }
    const float inv = 1.0f / red[0];
    __syncthreads();
    #pragma unroll
    for (int k = 0; k < 4; ++k)
        cv[tid + k * THREADS] = e[k] * inv;   // cv now holds a[b,:]
    __syncthreads();

    // ---- pass 2: g_out[b,c] = dot(l[b,c,:], a) ; l slice streamed from L2 ----
    const float4* a4 = (const float4*)cv;
    for (int c = wv; c < C_DIM; c += 8) {
        const float4* l4 = (const float4*)(lb + (size_t)c * HW_DIM);
        float s = 0.f;
        #pragma unroll
        for (int it = 0; it < 8; ++it) {
            const int idx = it * 32 + lane;
            float4 lv = l4[idx];
            float4 av = a4[idx];
            s += lv.x * av.x + lv.y * av.y + lv.z * av.z + lv.w * av.w;
        }
        #pragma unroll
        for (int off = 16; off > 0; off >>= 1)
            s += __shfl_xor(s, off, 32);
        if (lane == 0)
            out_g[b * C_DIM + c] = s;
    }
}

extern "C" void kernel_launch(void* const* d_in, const int* in_sizes, int n_in,
                              void* d_out, int out_size, void* d_ws, size_t ws_size,
                              hipStream_t stream) {
    const float* l = (const float*)d_in[0];   // [64, 512, 32, 32]
    const float* g = (const float*)d_in[1];   // [64, 512]
    const float* w = (const float*)d_in[2];   // [512]
    float* out   = (float*)d_out;
    float* out_c = out;                 // 64*1024 floats
    float* out_g = out + 64 * HW_DIM;   // 64*512 floats

    attn_fused_kernel<<<dim3(64), dim3(THREADS), 0, stream>>>(l, g, w, out_c, out_g);
}